// WSDDNOutputLayers_40389872451670
// MI455X (gfx1250) — compile-verified
//
#include <hip/hip_runtime.h>

typedef __attribute__((ext_vector_type(2))) float v2f;
typedef __attribute__((ext_vector_type(8))) float v8f;

constexpr int N_TOT = 131072;
constexpr int D     = 2048;
constexpr int K     = 20;
constexpr int KW    = 40;        // cls (20) + det (20) fused output columns
constexpr int NT    = 3;         // 48 cols = 3 WMMA 16-col tiles (last 8 dropped)
constexpr int BK    = 64;        // K-chunk per LDS stage
constexpr int ROWS  = 128;       // rows per block (8 waves x 16)
constexpr int LDX   = BK + 4;    // 68-float LDS row stride: 16B-aligned rows, conflict-free frag reads
constexpr int BOXD  = 80;        // NUM_BBOX_REG_CLASSES * 4

__global__ __launch_bounds__(256, 2)
void wsddn_wmma_kernel(const float* __restrict__ x,
                       const float* __restrict__ cls_w, const float* __restrict__ cls_b,
                       const float* __restrict__ det_w, const float* __restrict__ det_b,
                       float* __restrict__ out)
{
    __shared__ float xs[ROWS * LDX];   // x chunk: 128 x 64 (stride 68)
    __shared__ float wsm[48 * LDX];    // W chunk: 48 x 64 (rows 40..47 zero)

    const int tid     = threadIdx.x;
    const int wave    = tid >> 5;
    const int lane    = tid & 31;
    const int lrow    = lane & 15;
    const int halfSel = lane >> 4;     // 0: K=0,1 ; 1: K=2,3 (ISA A/B frag layout)
    const int rowBase = blockIdx.x * ROWS;
    const int r0      = wave * 16;

    // ---- zero the proposal_deltas slice for this block's rows (d_out is poisoned) ----
    {
        float4* del = reinterpret_cast<float4*>(out + (size_t)2 * N_TOT * K)
                      + (size_t)blockIdx.x * (ROWS * BOXD / 4);
        #pragma unroll
        for (int i = 0; i < ROWS * BOXD / 4 / 256; ++i)
            del[i * 256 + tid] = float4{0.f, 0.f, 0.f, 0.f};
    }

    // ---- zero pad rows 40..47 of W tile once (never overwritten) ----
    for (int i = tid; i < 8 * LDX; i += 256) wsm[40 * LDX + i] = 0.f;

    v8f acc0 = {}, acc1 = {}, acc2 = {};

    const float4* xg = reinterpret_cast<const float4*>(x);

    for (int kb = 0; kb < D; kb += BK) {
        // ---- stage x chunk: 128 rows x 16 float4, coalesced (16 threads/row = 256B) ----
        #pragma unroll
        for (int it = 0; it < 8; ++it) {
            int idx = it * 256 + tid;          // 0..2047
            int rr  = idx >> 4;
            int cc  = idx & 15;
            float4 v = xg[(size_t)(rowBase + rr) * (D / 4) + (kb >> 2) + cc];
            float* dst = &xs[rr * LDX + cc * 4];
            dst[0] = v.x; dst[1] = v.y; dst[2] = v.z; dst[3] = v.w;
        }
        // ---- stage W chunk: rows 0..19 = cls_w, 20..39 = det_w (640 float4) ----
        #pragma unroll
        for (int it = 0; it < 3; ++it) {
            int idx = it * 256 + tid;
            if (idx < KW * 16) {
                int rr = idx >> 4;
                int cc = idx & 15;
                const float4* wg = reinterpret_cast<const float4*>(rr < K ? cls_w : det_w);
                int wr = rr < K ? rr : rr - K;
                float4 v = wg[(size_t)wr * (D / 4) + (kb >> 2) + cc];
                float* dst = &wsm[rr * LDX + cc * 4];
                dst[0] = v.x; dst[1] = v.y; dst[2] = v.z; dst[3] = v.w;
            }
        }
        __syncthreads();

        // ---- prefetch next x chunk into caches (global_prefetch_b8) ----
        if (kb + BK < D) {
            const char* p = reinterpret_cast<const char*>(
                &x[(size_t)(rowBase + (tid >> 1)) * D + (kb + BK) + (tid & 1) * 32]);
            __builtin_prefetch(p, 0, 3);
        }

        // ---- 16 K-steps of V_WMMA_F32_16X16X4_F32, 3 column tiles each ----
        const float* aP = &xs[(r0 + lrow) * LDX + halfSel * 2];
        const float* b0 = &wsm[( 0 + lrow) * LDX + halfSel * 2];
        const float* b1 = &wsm[(16 + lrow) * LDX + halfSel * 2];
        const float* b2 = &wsm[(32 + lrow) * LDX + halfSel * 2];
        #pragma unroll
        for (int kk = 0; kk < BK / 4; ++kk) {
            v2f a   = *reinterpret_cast<const v2f*>(aP + kk * 4);
            v2f bb0 = *reinterpret_cast<const v2f*>(b0 + kk * 4);
            v2f bb1 = *reinterpret_cast<const v2f*>(b1 + kk * 4);
            v2f bb2 = *reinterpret_cast<const v2f*>(b2 + kk * 4);
            acc0 = __builtin_amdgcn_wmma_f32_16x16x4_f32(false, a, false, bb0, (short)0, acc0, false, false);
            acc1 = __builtin_amdgcn_wmma_f32_16x16x4_f32(false, a, false, bb1, (short)0, acc1, false, false);
            acc2 = __builtin_amdgcn_wmma_f32_16x16x4_f32(false, a, false, bb2, (short)0, acc2, false, false);
        }
        __syncthreads();
    }

    // ---- epilogue: C/D layout => each lane holds one column, rows M = r + 8*halfSel ----
    float* scoresO  = out;                          // (N, 20)
    float* weightsO = out + (size_t)N_TOT * K;      // (N, 20)

    v8f accs[NT] = {acc0, acc1, acc2};
    #pragma unroll
    for (int j = 0; j < NT; ++j) {
        int col = j * 16 + lrow;
        float bias; float* dstBase; int c; bool valid = true;
        if (col < K)            { bias = cls_b[col];     dstBase = scoresO;  c = col;     }
        else if (col < 2 * K)   { bias = det_b[col - K]; dstBase = weightsO; c = col - K; }
        else                    { bias = 0.f;            dstBase = scoresO;  c = 0; valid = false; }
        if (valid) {
            #pragma unroll
            for (int r = 0; r < 8; ++r) {
                int row = rowBase + r0 + r + halfSel * 8;
                dstBase[(size_t)row * K + c] = accs[j][r] + bias;
            }
        }
    }
}

extern "C" void kernel_launch(void* const* d_in, const int* in_sizes, int n_in,
                              void* d_out, int out_size, void* d_ws, size_t ws_size,
                              hipStream_t stream) {
    (void)in_sizes; (void)n_in; (void)d_ws; (void)ws_size; (void)out_size;
    const float* x     = (const float*)d_in[0];
    const float* cls_w = (const float*)d_in[1];
    const float* cls_b = (const float*)d_in[2];
    const float* det_w = (const float*)d_in[3];
    const float* det_b = (const float*)d_in[4];
    float* out = (float*)d_out;

    dim3 grid(N_TOT / ROWS);   // 1024 blocks
    dim3 block(256);           // 8 wave32
    wsddn_wmma_kernel<<<grid, block, 0, stream>>>(x, cls_w, cls_b, det_w, det_b, out);
}